// SAGE_29609504539563
// MI455X (gfx1250) — compile-verified
//
#include <hip/hip_runtime.h>
#include <hip/hip_bf16.h>

#define DFEAT 128

typedef float v2f __attribute__((ext_vector_type(2)));
typedef float v8f __attribute__((ext_vector_type(8)));

__device__ __forceinline__ v8f wmma_f32(v2f a, v2f b, v8f c) {
    // D = A(16x4) * B(4x16) + C(16x16), fp32 throughout
    return __builtin_amdgcn_wmma_f32_16x16x4_f32(
        /*neg_a=*/false, a, /*neg_b=*/false, b,
        /*c_mod=*/(short)0, c, /*reuse_a=*/false, /*reuse_b=*/false);
}

// ---------------------------------------------------------------------------
// Edge scatter: agg[dst] += feat[src]; cnt[dst] += 1 (optional).
// One wave32 per edge; each lane moves one float4 (128 floats / 32 lanes),
// accumulated with native non-returning global_atomic_add_f32.
// ---------------------------------------------------------------------------
__global__ void sage_scatter(const float* __restrict__ feat,
                             const int* __restrict__ src,
                             const int* __restrict__ dst,
                             float* __restrict__ agg,
                             float* __restrict__ cnt,
                             int E, int addCnt) {
    int gid  = blockIdx.x * blockDim.x + threadIdx.x;
    int wid  = gid >> 5;
    int lane = threadIdx.x & 31;
    if (wid >= E) return;
    int s = src[wid];
    int d = dst[wid];
    const float4 v = ((const float4*)(feat + (size_t)s * DFEAT))[lane];
    float* dp = agg + (size_t)d * DFEAT + lane * 4;
    unsafeAtomicAdd(dp + 0, v.x);
    unsafeAtomicAdd(dp + 1, v.y);
    unsafeAtomicAdd(dp + 2, v.z);
    unsafeAtomicAdd(dp + 3, v.w);
    if (addCnt && lane == 0) unsafeAtomicAdd(cnt + d, 1.0f);
}

// ---------------------------------------------------------------------------
// Fused SAGE layer: out = act( (agg/max(cnt,1)) @ Wl + xin @ Wr + bias )
// Persistent blocks of 256 threads (8 waves), grid-stride over 16-row tiles.
// Weights live in 128 KB LDS, pre-swizzled into the WMMA B-register layout:
//   lds[(r>>2)*1024 + mat*512 + c*4 + (r&3)] = W[r][c]
// so each lane's B operand is ONE contiguous ds_load_b64 at a small constant
// dword offset (t*64 / 512+t*64) off a single per-kstep base register.
// ---------------------------------------------------------------------------
__global__ void sage_gemm(const float* __restrict__ agg,
                          const float* __restrict__ cnt,
                          const float* xin,
                          const float* __restrict__ Wl,
                          const float* __restrict__ Wr,
                          const float* __restrict__ bias,
                          float* out,
                          int doRelu, int nrows) {
    extern __shared__ float lds[];                 // 2 * 128 * 128 floats
    const int tid = threadIdx.x;                   // 0..255

    // Stage both weight matrices, swizzling into B-register order.
    for (int i = tid; i < 4096; i += 256) {        // 4096 float4 per matrix
        const int r  = i >> 5;                     // row 0..127
        const int c4 = (i & 31) << 2;              // col 0,4,..,124
        float* p = lds + ((r >> 2) << 10) + (r & 3) + (c4 << 2);
        const float4 vl = ((const float4*)Wl)[i];
        p[0] = vl.x; p[4] = vl.y; p[8] = vl.z; p[12] = vl.w;
        const float4 vr = ((const float4*)Wr)[i];
        p[512] = vr.x; p[516] = vr.y; p[520] = vr.z; p[524] = vr.w;
    }
    __syncthreads();

    const int wave  = tid >> 5;                    // 0..7
    const int lane  = tid & 31;
    const int m     = lane & 15;                   // A row / B,C col
    const int half  = lane >> 4;
    const int khalf = half << 1;                   // 0 or 2
    const int ntile = nrows >> 4;
    const float* bbase = lds + (m << 2) + khalf;   // lane-constant B base

#pragma unroll 1
    for (int tile = blockIdx.x * 8 + wave; tile < ntile; tile += gridDim.x * 8) {
        const int row   = tile * 16 + m;
        const float cvl = cnt[row];
        const float inv = 1.0f / fmaxf(cvl, 1.0f);
        const float* aggRow = agg + (size_t)row * DFEAT;
        const float* xRow   = xin + (size_t)row * DFEAT;

        v8f acc[8];
#pragma unroll
        for (int t = 0; t < 8; ++t)
#pragma unroll
            for (int j = 0; j < 8; ++j) acc[t][j] = 0.0f;

        for (int ks = 0; ks < 32; ++ks) {
            const int k = ks * 4 + khalf;
            v2f a1 = *(const v2f*)(aggRow + k);    // global_load_b64
            a1.x *= inv;
            a1.y *= inv;
            const v2f a2 = *(const v2f*)(xRow + k);
            const float* bk = bbase + (ks << 10);  // one add per kstep
#pragma unroll
            for (int t = 0; t < 8; ++t) {
                const v2f bl = *(const v2f*)(bk + t * 64);        // ds_load_b64
                acc[t] = wmma_f32(a1, bl, acc[t]);
                const v2f br = *(const v2f*)(bk + 512 + t * 64);  // ds_load_b64
                acc[t] = wmma_f32(a2, br, acc[t]);
            }
        }

        // Epilogue: bias (+ReLU), store. C/D: col = m, row = j + 8*half.
#pragma unroll
        for (int t = 0; t < 8; ++t) {
            const float bv = bias[t * 16 + m];
#pragma unroll
            for (int j = 0; j < 8; ++j) {
                const int r = tile * 16 + j + 8 * half;
                float v = acc[t][j] + bv;
                if (doRelu) v = fmaxf(v, 0.0f);
                out[(size_t)r * DFEAT + t * 16 + m] = v;
            }
        }
    }
}

// ---------------------------------------------------------------------------
// Pair scorer: per 16 pairs, A = concat(h[p0], h[p1]) (16x256),
// z = relu(A @ Ws1 + bs1) via WMMA (4 accumulators = 64 cols),
// score = z @ Ws2 + bs2 via in-register dot + 16-lane shfl reduction,
// out = sigmoid(score). Ws1 staged in LDS in the same swizzled B layout:
//   lds[(r>>2)*256 + c*4 + (r&3)] = Ws1[r][c]   (r<256, c<64)
// ---------------------------------------------------------------------------
__global__ void sage_pairs(const float* __restrict__ h,
                           const int* __restrict__ pairs,
                           const float* __restrict__ Ws1,
                           const float* __restrict__ bsws,  // packed bs1|Ws2
                           const float* __restrict__ bs2,
                           float* __restrict__ out, int P) {
    extern __shared__ float lds[];                 // 256 * 64 floats
    const int tid = threadIdx.x;                   // 0..255
    for (int i = tid; i < 4096; i += 256) {        // 4096 float4 total
        const int r  = i >> 4;                     // row 0..255
        const int c4 = (i & 15) << 2;              // col 0,4,..,60
        float* p = lds + ((r >> 2) << 8) + (r & 3) + (c4 << 2);
        const float4 v = ((const float4*)Ws1)[i];
        p[0] = v.x; p[4] = v.y; p[8] = v.z; p[12] = v.w;
    }
    __syncthreads();

    const int wave  = tid >> 5;
    const int lane  = tid & 31;
    const int m     = lane & 15;
    const int half  = lane >> 4;
    const int khalf = half << 1;
    const int ntile = P >> 4;
    const float* bbase = lds + (m << 2) + khalf;

    // Per-lane epilogue constants (cached, tiny).
    float bv[4], wv[4];
#pragma unroll
    for (int t = 0; t < 4; ++t) {
        bv[t] = bsws[t * 16 + m];                  // bs1[col]
        wv[t] = bsws[64 + t * 16 + m];             // Ws2[col]
    }
    const float b2v = bs2[0];

#pragma unroll 1
    for (int tile = blockIdx.x * 8 + wave; tile < ntile; tile += gridDim.x * 8) {
        const int pidx = tile * 16 + m;
        const int p0 = pairs[pidx];
        const int p1 = pairs[P + pidx];
        const float* rowA = h + (size_t)p0 * DFEAT;
        const float* rowB = h + (size_t)p1 * DFEAT;

        v8f acc[4];
#pragma unroll
        for (int t = 0; t < 4; ++t)
#pragma unroll
            for (int j = 0; j < 8; ++j) acc[t][j] = 0.0f;

#pragma unroll 1
        for (int hs = 0; hs < 2; ++hs) {           // K halves: h[p0], h[p1]
            const float* rsrc = hs ? rowB : rowA;
            for (int ks = 0; ks < 32; ++ks) {
                const int kk = ks * 4 + khalf;     // 0..127 within half
                const v2f a = *(const v2f*)(rsrc + kk);
                const float* bk = bbase + ((hs * 32 + ks) << 8);
#pragma unroll
                for (int t = 0; t < 4; ++t) {
                    const v2f b = *(const v2f*)(bk + t * 64);  // ds_load_b64
                    acc[t] = wmma_f32(a, b, acc[t]);
                }
            }
        }

        // z = relu(acc + bs1); partial[j] = sum_cols z * Ws2[col]
        float partial[8];
#pragma unroll
        for (int j = 0; j < 8; ++j) partial[j] = 0.0f;
#pragma unroll
        for (int t = 0; t < 4; ++t)
#pragma unroll
            for (int j = 0; j < 8; ++j) {
                const float z = fmaxf(acc[t][j] + bv[t], 0.0f);
                partial[j] += z * wv[t];
            }
        // Reduce across the 16 lanes of this half-wave (cols 0..15).
#pragma unroll
        for (int off = 8; off >= 1; off >>= 1)
#pragma unroll
            for (int j = 0; j < 8; ++j)
                partial[j] += __shfl_xor(partial[j], off, 32);

        if (m == 0) {
#pragma unroll
            for (int j = 0; j < 8; ++j) {
                const float s = partial[j] + b2v;  // row = j + 8*half
                out[(size_t)tile * 16 + 8 * half + j] = 1.0f / (1.0f + __expf(-s));
            }
        }
    }
}

// Pack bs1 (64) and Ws2 (64) contiguously so the pair kernel reads both
// through one cached pointer.
__global__ void pack_bs(const float* __restrict__ bs1,
                        const float* __restrict__ Ws2,
                        float* __restrict__ dstbuf) {
    int t = threadIdx.x;
    if (t < 64) dstbuf[t] = bs1[t];
    else if (t < 128) dstbuf[t] = Ws2[t - 64];
}

// ---------------------------------------------------------------------------
extern "C" void kernel_launch(void* const* d_in, const int* in_sizes, int n_in,
                              void* d_out, int out_size, void* d_ws, size_t ws_size,
                              hipStream_t stream) {
    const float* x    = (const float*)d_in[0];
    const int*   eix  = (const int*)  d_in[1];
    const int*   prs  = (const int*)  d_in[2];
    const float* W1l  = (const float*)d_in[3];
    const float* W1r  = (const float*)d_in[4];
    const float* b1   = (const float*)d_in[5];
    const float* W2l  = (const float*)d_in[6];
    const float* W2r  = (const float*)d_in[7];
    const float* b2   = (const float*)d_in[8];
    const float* Ws1  = (const float*)d_in[9];
    const float* bs1  = (const float*)d_in[10];
    const float* Ws2  = (const float*)d_in[11];
    const float* bs2  = (const float*)d_in[12];

    const int N = in_sizes[0] / DFEAT;   // 100000
    const int E = in_sizes[1] / 2;       // 1.6M
    const int P = in_sizes[2] / 2;       // 500K

    // Workspace: agg (N*128) | h (N*128) | cnt (N) | packed bs1,Ws2 (128).
    // Layer-2 GEMM writes h in place (each wave exclusively owns its 16 rows).
    float* agg  = (float*)d_ws;
    float* hbf  = agg + (size_t)N * DFEAT;
    float* cnt  = hbf + (size_t)N * DFEAT;
    float* bsws = cnt + N;

    const int* src = eix;
    const int* dst = eix + E;

    const int scatterBlocks = (int)(((long long)E * 32 + 255) / 256);
    const int gemmTiles = N / 16, pairTiles = P / 16;
    int gemmBlocks = 256;                          // persistent: 1 per WGP-ish
    if (gemmBlocks > (gemmTiles + 7) / 8) gemmBlocks = (gemmTiles + 7) / 8;
    int pairBlocks = 512;
    if (pairBlocks > (pairTiles + 7) / 8) pairBlocks = (pairTiles + 7) / 8;
    const size_t gemmLds = 2u * 128u * 128u * sizeof(float);      // 128 KB
    const size_t pairLds = 256u * 64u * sizeof(float);            // 64 KB

    pack_bs<<<1, 128, 0, stream>>>(bs1, Ws2, bsws);

    // ---- Layer 1 ----
    hipMemsetAsync(agg, 0, (size_t)N * DFEAT * sizeof(float), stream);
    hipMemsetAsync(cnt, 0, (size_t)N * sizeof(float), stream);
    sage_scatter<<<scatterBlocks, 256, 0, stream>>>(x, src, dst, agg, cnt, E, 1);
    sage_gemm<<<gemmBlocks, 256, gemmLds, stream>>>(agg, cnt, x, W1l, W1r, b1,
                                                    hbf, /*relu=*/1, N);

    // ---- Layer 2 (degree counts reused) ----
    hipMemsetAsync(agg, 0, (size_t)N * DFEAT * sizeof(float), stream);
    sage_scatter<<<scatterBlocks, 256, 0, stream>>>(hbf, src, dst, agg, cnt, E, 0);
    sage_gemm<<<gemmBlocks, 256, gemmLds, stream>>>(agg, cnt, hbf, W2l, W2r, b2,
                                                    hbf, /*relu=*/0, N);

    // ---- Pair scoring ----
    sage_pairs<<<pairBlocks, 256, pairLds, stream>>>(hbf, prs, Ws1, bsws, bs2,
                                                     (float*)d_out, P);
}